// VAELoss_11965778886768
// MI455X (gfx1250) — compile-verified
//
#include <hip/hip_runtime.h>
#include <hip/hip_bf16.h>
#include <math.h>

typedef __attribute__((ext_vector_type(2))) float v2f;
typedef __attribute__((ext_vector_type(8))) float v8f;

#define CB 8
#define CN 4096
#define CM 4096
#define CL 256
#define CV 64
#define INF_F 3.0e38f

// ---------------------------------------------------------------------------
// ws layout (floats): [0]=rowmin-sum pass1, [1]=rowmin-sum pass2, [2]=KL raw
// sum, [3]=BCE sum, [4]=smooth sum
// ---------------------------------------------------------------------------
__global__ void vae_init_ws(float* ws) {
    if (threadIdx.x < 8) ws[threadIdx.x] = 0.0f;
}

// ---------------------------------------------------------------------------
// Chamfer one-directional: for each row point i, min_j ||r_i - c_j||^2 ;
// accumulates sum of row-mins into *accum. One wave per 16-row tile.
// Uses v_wmma_f32_16x16x4_f32 with K padded 3->4.
// ---------------------------------------------------------------------------
__global__ __launch_bounds__(256) void chamfer_rowmin_kernel(
    const float* __restrict__ rows,   // [B, NR, 3]
    const float* __restrict__ cols,   // [B, NC, 3]
    int NR, int NC,
    float* __restrict__ accum)
{
    const int lane = threadIdx.x & 31;
    const int wave = threadIdx.x >> 5;
    const int tile = blockIdx.x * 8 + wave;       // global 16-row tile id
    const int tilesPerBatch = NR >> 4;
    const int b     = tile / tilesPerBatch;
    const int rbase = (tile % tilesPerBatch) << 4;

    const float* rp = rows + ((size_t)b * NR + rbase) * 3;
    const float* cp = cols + (size_t)b * NC * 3;

    const int  l16 = lane & 15;
    const bool hi  = lane >= 16;

    // A tile (16x4, K padded): low half lanes carry {x,y}, high half {z,0}
    const float px = rp[l16 * 3 + 0];
    const float py = rp[l16 * 3 + 1];
    const float pz = rp[l16 * 3 + 2];
    v2f a;
    a[0] = hi ? pz : px;
    a[1] = hi ? 0.0f : py;
    const float pn = px * px + py * py + pz * pz;  // ||r_i||^2, row = l16

    float acc[8];
#pragma unroll
    for (int v = 0; v < 8; ++v) acc[v] = INF_F;

    for (int mb = 0; mb < NC; mb += 16) {
        const float* tp = cp + (size_t)(mb + l16) * 3;
        const float tx = tp[0], ty = tp[1], tz = tp[2];
        const float gn = tx * tx + ty * ty + tz * tz;  // ||c_j||^2, col = l16
        v2f bv;
        bv[0] = hi ? tz : tx;
        bv[1] = hi ? 0.0f : ty;
        v8f c = {};
        c = __builtin_amdgcn_wmma_f32_16x16x4_f32(
                /*neg_a=*/false, a, /*neg_b=*/false, bv,
                /*c_mod=*/(short)0, c, /*reuse_a=*/false, /*reuse_b=*/false);
#pragma unroll
        for (int v = 0; v < 8; ++v) {
            const float d = gn - 2.0f * c[v];    // dist - pn_i (row-constant)
            acc[v] = fminf(acc[v], d);
        }
    }

    // min across the 16 lanes of each half (covers all columns)
#pragma unroll
    for (int v = 0; v < 8; ++v) {
        float x = acc[v];
        x = fminf(x, __shfl_xor(x, 1, 32));
        x = fminf(x, __shfl_xor(x, 2, 32));
        x = fminf(x, __shfl_xor(x, 4, 32));
        x = fminf(x, __shfl_xor(x, 8, 32));
        acc[v] = x;
    }
    float s = 0.0f;
#pragma unroll
    for (int v = 0; v < 8; ++v) s += acc[v];     // half0: rows 0-7, half1: 8-15

    // sum of pn over the 16 rows (each half holds all 16 rows once)
    float ps = pn;
    ps += __shfl_xor(ps, 1, 32);
    ps += __shfl_xor(ps, 2, 32);
    ps += __shfl_xor(ps, 4, 32);
    ps += __shfl_xor(ps, 8, 32);

    const float sOther = __shfl_xor(s, 16, 32);
    if (lane == 0) atomicAdd(accum, s + sOther + ps);
}

// ---------------------------------------------------------------------------
// KL raw sum: sum(1 + logvar - mu^2 - exp(logvar))
// ---------------------------------------------------------------------------
__global__ __launch_bounds__(256) void kl_kernel(
    const float* __restrict__ mu, const float* __restrict__ logvar,
    int n, float* __restrict__ accum)
{
    __shared__ float red;
    if (threadIdx.x == 0) red = 0.0f;
    __syncthreads();
    float local = 0.0f;
    for (int i = blockIdx.x * blockDim.x + threadIdx.x; i < n;
         i += gridDim.x * blockDim.x) {
        const float m = mu[i], lv = logvar[i];
        local += 1.0f + lv - m * m - expf(lv);
    }
    atomicAdd(&red, local);
    __syncthreads();
    if (threadIdx.x == 0) atomicAdd(accum, red);
}

// ---------------------------------------------------------------------------
// Voxel BCE sum: -(t*log p + (1-t)*log1p(-p)), p clipped to [EPS, 1-EPS]
// ---------------------------------------------------------------------------
__global__ __launch_bounds__(256) void bce_kernel(
    const float* __restrict__ vox, const float* __restrict__ tvox,
    int n, float* __restrict__ accum)
{
    __shared__ float red;
    if (threadIdx.x == 0) red = 0.0f;
    __syncthreads();
    const float EPS = 1e-7f;
    float local = 0.0f;
    for (int i = blockIdx.x * blockDim.x + threadIdx.x; i < n;
         i += gridDim.x * blockDim.x) {
        float p = vox[i];
        p = fminf(fmaxf(p, EPS), 1.0f - EPS);
        const float t = tvox[i];
        local -= t * logf(p) + (1.0f - t) * log1pf(-p);
    }
    atomicAdd(&red, local);
    __syncthreads();
    if (threadIdx.x == 0) atomicAdd(accum, red);
}

// ---------------------------------------------------------------------------
// Smoothness: per (b, coord) bitonic-sort 4096 floats in LDS, then sum of
// squared consecutive diffs.
// ---------------------------------------------------------------------------
__global__ __launch_bounds__(256) void smooth_kernel(
    const float* __restrict__ pts, float* __restrict__ accum)
{
    __shared__ float s[CN];
    __shared__ float red;
    const int b = blockIdx.x / 3;
    const int c = blockIdx.x % 3;
    const float* p = pts + (size_t)b * CN * 3 + c;
    for (int i = threadIdx.x; i < CN; i += 256) s[i] = p[(size_t)i * 3];
    if (threadIdx.x == 0) red = 0.0f;
    __syncthreads();

    for (int k = 2; k <= CN; k <<= 1) {
        for (int j = k >> 1; j > 0; j >>= 1) {
            for (int i = threadIdx.x; i < CN; i += 256) {
                const int ixj = i ^ j;
                if (ixj > i) {
                    const bool up = ((i & k) == 0);
                    const float x = s[i], y = s[ixj];
                    if (up ? (x > y) : (x < y)) { s[i] = y; s[ixj] = x; }
                }
            }
            __syncthreads();
        }
    }

    float local = 0.0f;
    for (int i = threadIdx.x; i < CN - 1; i += 256) {
        const float d = s[i + 1] - s[i];
        local += d * d;
    }
    atomicAdd(&red, local);
    __syncthreads();
    if (threadIdx.x == 0) atomicAdd(accum, red);
}

// ---------------------------------------------------------------------------
// Finalize: assemble the 5 outputs with reference weights.
// ---------------------------------------------------------------------------
__global__ void finalize_kernel(const float* __restrict__ ws,
                                float* __restrict__ out)
{
    if (threadIdx.x != 0 || blockIdx.x != 0) return;
    const float point = ws[0] / (float)(CB * CN) + ws[1] / (float)(CB * CM);
    const float kl    = -0.5f * ws[2];
    const float vox   = ws[3] / (float)(CB * CV * CV * CV);
    const float sm    = ws[4] / (float)(CB * (CN - 1));
    const float total = 1.0f * point + 0.001f * kl + 0.5f * vox + 0.1f * sm;
    out[0] = total;
    out[1] = point;
    out[2] = kl;
    out[3] = vox;
    out[4] = sm;
}

extern "C" void kernel_launch(void* const* d_in, const int* in_sizes, int n_in,
                              void* d_out, int out_size, void* d_ws, size_t ws_size,
                              hipStream_t stream) {
    (void)in_sizes; (void)n_in; (void)out_size; (void)ws_size;
    const float* pred   = (const float*)d_in[0];  // [8,4096,3]
    const float* tgt    = (const float*)d_in[1];  // [8,4096,3]
    const float* mu     = (const float*)d_in[2];  // [8,256]
    const float* logvar = (const float*)d_in[3];  // [8,256]
    const float* vox    = (const float*)d_in[4];  // [8,64,64,64]
    const float* tvox   = (const float*)d_in[5];  // [8,64,64,64]
    float* out = (float*)d_out;
    float* ws  = (float*)d_ws;

    vae_init_ws<<<1, 32, 0, stream>>>(ws);

    // chamfer: 2048 row-tiles per direction, 8 waves/block -> 256 blocks
    chamfer_rowmin_kernel<<<256, 256, 0, stream>>>(pred, tgt, CN, CM, ws + 0);
    chamfer_rowmin_kernel<<<256, 256, 0, stream>>>(tgt, pred, CM, CN, ws + 1);

    kl_kernel<<<8, 256, 0, stream>>>(mu, logvar, CB * CL, ws + 2);
    bce_kernel<<<1024, 256, 0, stream>>>(vox, tvox, CB * CV * CV * CV, ws + 3);
    smooth_kernel<<<CB * 3, 256, 0, stream>>>(pred, ws + 4);

    finalize_kernel<<<1, 32, 0, stream>>>(ws, out);
}